// GNN_Cluster_42425686950039
// MI455X (gfx1250) — compile-verified
//
#include <hip/hip_runtime.h>
#include <hip/hip_bf16.h>

typedef __attribute__((ext_vector_type(2))) float v2f;
typedef __attribute__((ext_vector_type(8))) float v8f;

#define TM 128
#define TN 128
#define TK 32
#define LDSP (TK + 2)   // padded LDS row stride (dwords); keeps 8B alignment, bank-conflict-free

// ---------------------------------------------------------------------------
// Generic fp32 WMMA GEMM:  C = A @ B (+ beta*C), optional B-transpose, optional ReLU.
// Block = 256 threads (8 waves). Macro-tile 128x128, K staged in chunks of 32.
// Wave w computes rows [16w,16w+16) x all 128 cols => 8 16x16 accumulators.
// A fragment (16x4): lane L -> M = L%16, holds K = k0+2*(L/16) .. +1   (2 VGPRs)
// B fragment (4x16): lane L -> N = L%16, holds K = k0+2*(L/16) .. +1   (2 VGPRs)
// ---------------------------------------------------------------------------
__global__ void __launch_bounds__(256)
wmma_gemm_f32(float* __restrict__ C, const float* __restrict__ A,
              const float* __restrict__ B,
              int M, int Nn, int K, int lda, int ldb, int ldc,
              float beta, int transB, int relu)
{
    __shared__ float sA[TM * LDSP];   // sA[row][k]
    __shared__ float sB[TN * LDSP];   // sB[col][k]  (B stored transposed)

    const int tid  = threadIdx.x;
    const int lane = tid & 31;
    const int wave = tid >> 5;
    const int mloc = lane & 15;       // M (or N) index within 16
    const int g    = lane >> 4;       // half-wave selector

    const int rowBase = blockIdx.y * TM;
    const int colBase = blockIdx.x * TN;

    v8f acc[8] = {};

    for (int kk = 0; kk < K; kk += TK) {
        // ---- stage A: 128x32 tile, float4 loads ----
        #pragma unroll
        for (int i = 0; i < 4; ++i) {
            int u  = tid + 256 * i;            // 0..1023
            int r  = u >> 3;                   // 0..127
            int c4 = (u & 7) << 2;             // 0,4,..,28
            const float* gp = A + (size_t)(rowBase + r) * lda + kk + c4;
            float4 v = *(const float4*)gp;
            if (kk + TK < K) __builtin_prefetch(gp + TK, 0, 0);
            sA[r * LDSP + c4 + 0] = v.x;
            sA[r * LDSP + c4 + 1] = v.y;
            sA[r * LDSP + c4 + 2] = v.z;
            sA[r * LDSP + c4 + 3] = v.w;
        }
        // ---- stage B (transposed into LDS) ----
        if (!transB) {
            #pragma unroll
            for (int i = 0; i < 16; ++i) {
                int u = tid + 256 * i;         // 0..4095
                int k = u >> 7;                // 0..31
                int c = u & 127;               // 0..127  (coalesced along c)
                sB[c * LDSP + k] = B[(size_t)(kk + k) * ldb + colBase + c];
            }
        } else {
            #pragma unroll
            for (int i = 0; i < 16; ++i) {
                int u = tid + 256 * i;
                int c = u >> 5;                // 0..127
                int k = u & 31;                // 0..31   (coalesced along k)
                sB[c * LDSP + k] = B[(size_t)(colBase + c) * ldb + kk + k];
            }
        }
        __syncthreads();

        // ---- 8 k-steps x 8 n-tiles = 64 WMMAs per stage ----
        #pragma unroll
        for (int k0 = 0; k0 < TK; k0 += 4) {
            const int ka = k0 + 2 * g;
            v2f a;
            a.x = sA[(wave * 16 + mloc) * LDSP + ka];
            a.y = sA[(wave * 16 + mloc) * LDSP + ka + 1];
            #pragma unroll
            for (int nt = 0; nt < 8; ++nt) {
                v2f b;
                b.x = sB[(nt * 16 + mloc) * LDSP + ka];
                b.y = sB[(nt * 16 + mloc) * LDSP + ka + 1];
                acc[nt] = __builtin_amdgcn_wmma_f32_16x16x4_f32(
                    false, a, false, b, (short)0, acc[nt], false, false);
            }
        }
        __syncthreads();
    }

    // ---- epilogue: C/D layout -> row = wave*16 + r + 8*g, col = nt*16 + mloc
    #pragma unroll
    for (int nt = 0; nt < 8; ++nt) {
        #pragma unroll
        for (int r = 0; r < 8; ++r) {
            int row = rowBase + wave * 16 + r + 8 * g;
            int col = colBase + nt * 16 + mloc;
            size_t idx = (size_t)row * ldc + col;
            float v = acc[nt][r];
            if (beta != 0.0f) v += beta * C[idx];
            if (relu) v = fmaxf(v, 0.0f);
            C[idx] = v;
        }
    }
}

// ---------------------------------------------------------------------------
// Elementwise / scatter helpers
// ---------------------------------------------------------------------------
__global__ void k_zero(float* p, int n) {
    for (int i = blockIdx.x * blockDim.x + threadIdx.x; i < n; i += gridDim.x * blockDim.x)
        p[i] = 0.0f;
}

__global__ void k_scatter_add(const int* __restrict__ src, const int* __restrict__ tgt,
                              const float* __restrict__ feat, float* __restrict__ agg,
                              int E, int H) {
    int i = blockIdx.x * blockDim.x + threadIdx.x;
    if (i >= E * H) return;
    int e = i / H, d = i % H;
    atomicAdd(&agg[(size_t)tgt[e] * H + d], feat[(size_t)src[e] * H + d]);
}

__global__ void k_count(const int* __restrict__ tgt, float* __restrict__ cnt, int E) {
    int e = blockIdx.x * blockDim.x + threadIdx.x;
    if (e < E) atomicAdd(&cnt[tgt[e]], 1.0f);
}

__global__ void k_divcnt(float* agg, const float* __restrict__ cnt, int N, int H) {
    int i = blockIdx.x * blockDim.x + threadIdx.x;
    if (i < N * H) agg[i] /= fmaxf(cnt[i / H], 1.0f);
}

__global__ void k_bias3(float* out, const float* __restrict__ b, int N, int H) {
    int i = blockIdx.x * blockDim.x + threadIdx.x;
    if (i < N * H) {
        int d = i % H;
        out[i] = b[d] + b[H + d] + b[2 * H + d];
    }
}

__global__ void k_relu(float* p, int n) {
    int i = blockIdx.x * blockDim.x + threadIdx.x;
    if (i < n) p[i] = fmaxf(p[i], 0.0f);
}

__global__ void k_rowq(const float* __restrict__ z, const float* __restrict__ w,
                       float* __restrict__ zw, float* __restrict__ q, int N, int H) {
    int r = blockIdx.x * blockDim.x + threadIdx.x;
    if (r >= N) return;
    float s = 0.0f;
    for (int d = 0; d < H; ++d) {
        float zz = z[(size_t)r * H + d];
        float ww = w[d];
        zw[(size_t)r * H + d] = zz * ww;
        s += zz * zz * ww;
    }
    q[r] = s;
}

// G (= zw@z^T) -> sym = sigmoid(q_i + q_j - 2G + lin_b), doubled diagonal
__global__ void k_sym(float* G, const float* __restrict__ q,
                      const float* __restrict__ lin_b, int N) {
    int i = blockIdx.x * blockDim.x + threadIdx.x;
    if (i >= N * N) return;
    int r = i / N, c = i % N;
    float x = q[r] + q[c] - 2.0f * G[i] + lin_b[0];
    float s = 1.0f / (1.0f + expf(-x));
    if (r == c) s *= 2.0f;
    G[i] = s;
}

__global__ void k_sumsq(const float* __restrict__ p, float* out, int n) {
    __shared__ float red[256];
    int tid = threadIdx.x;
    float s = 0.0f;
    for (int i = blockIdx.x * blockDim.x + tid; i < n; i += gridDim.x * blockDim.x)
        s += p[i] * p[i];
    red[tid] = s;
    __syncthreads();
    for (int off = 128; off > 0; off >>= 1) {
        if (tid < off) red[tid] += red[tid + off];
        __syncthreads();
    }
    if (tid == 0) atomicAdd(out, red[0]);
}

__global__ void k_initYZ(const float* __restrict__ A, const float* __restrict__ sumsq,
                         float* __restrict__ Y, float* __restrict__ Z, int N) {
    int i = blockIdx.x * blockDim.x + threadIdx.x;
    if (i >= N * N) return;
    float nrm = sqrtf(sumsq[0]);
    Y[i] = A[i] / nrm;
    int r = i / N, c = i % N;
    Z[i] = (r == c) ? 1.0f : 0.0f;
}

// Tm (holding Z@Y) -> 0.5*(3I - Z@Y)
__global__ void k_tm(float* Tm, int N) {
    int i = blockIdx.x * blockDim.x + threadIdx.x;
    if (i >= N * N) return;
    int r = i / N, c = i % N;
    Tm[i] = 0.5f * ((r == c ? 3.0f : 0.0f) - Tm[i]);
}

__global__ void k_scaleS1(const float* __restrict__ Y, const float* __restrict__ sumsq,
                          float* __restrict__ S1, int n) {
    int i = blockIdx.x * blockDim.x + threadIdx.x;
    if (i < n) S1[i] = Y[i] * sqrtf(sqrtf(sumsq[0]));   // * sqrt(normA), normA = sqrt(sumsq)
}

__global__ void k_setedges(const int* __restrict__ src, const int* __restrict__ tgt,
                           float* __restrict__ adj, int N, int E) {
    int e = blockIdx.x * blockDim.x + threadIdx.x;
    if (e < E) adj[(size_t)src[e] * N + tgt[e]] = 1.0f;
}

// ---------------------------------------------------------------------------
extern "C" void kernel_launch(void* const* d_in, const int* in_sizes, int n_in,
                              void* d_out, int out_size, void* d_ws, size_t ws_size,
                              hipStream_t stream) {
    (void)n_in; (void)out_size; (void)ws_size;
    const int H = 128;
    const int N = in_sizes[0] / H;      // 2048
    const int E = in_sizes[1] / 2;      // 32768

    const float* x   = (const float*)d_in[0];
    const int*   ei[3] = {(const int*)d_in[1], (const int*)d_in[2], (const int*)d_in[3]};
    const float* W1s = (const float*)d_in[4];
    const float* W1n = (const float*)d_in[5];
    const float* b1  = (const float*)d_in[6];
    const float* W2s = (const float*)d_in[7];
    const float* W2n = (const float*)d_in[8];
    const float* b2  = (const float*)d_in[9];
    const float* lw  = (const float*)d_in[10];
    const float* lb  = (const float*)d_in[11];

    float* out    = (float*)d_out;
    float* o_xnew = out;                                   // N*H
    float* o_A1   = out + (size_t)N * H;                   // N*N
    float* o_A2   = o_A1 + (size_t)N * N;
    float* o_A3   = o_A2 + (size_t)N * N;
    float* o_S1   = o_A3 + (size_t)N * N;
    float* o_A[3] = {o_A1, o_A2, o_A3};

    float* w = (float*)d_ws;
    float* agg   = w;  w += (size_t)N * H;
    float* cnt   = w;  w += (size_t)N;
    float* hbuf  = w;  w += (size_t)N * H;
    float* zbuf  = w;  w += (size_t)N * H;
    float* zwb   = w;  w += (size_t)N * H;
    float* qv    = w;  w += (size_t)N;
    float* sumsq = w;  w += 16;
    float* symTm = w;  w += (size_t)N * N;
    float* Y0    = w;  w += (size_t)N * N;
    float* Y1    = w;  w += (size_t)N * N;
    float* Z0    = w;  w += (size_t)N * N;
    float* Z1    = w;  w += (size_t)N * N;

    const int NN  = N * N;
    const int NH  = N * H;
    dim3 b256(256);
    auto blk = [](int n) { return dim3((unsigned)((n + 255) / 256)); };

    auto gemm = [&](float* C, const float* A, const float* B,
                    int M, int Nn, int K, int lda, int ldb, int ldc,
                    float beta, int transB, int relu) {
        dim3 grid(Nn / TN, M / TM);
        wmma_gemm_f32<<<grid, b256, 0, stream>>>(C, A, B, M, Nn, K, lda, ldb, ldc,
                                                 beta, transB, relu);
    };

    // ---------------- hetero SAGE layer ----------------
    auto layer = [&](float* ob, const float* ib, const float* Ws, const float* Wn,
                     const float* bb, int relu_last) {
        k_bias3<<<blk(NH), b256, 0, stream>>>(ob, bb, N, H);
        for (int t = 0; t < 3; ++t) {
            gemm(ob, ib, Ws + (size_t)t * H * H, N, H, H, H, H, H, 1.0f, 0, 0);
            k_zero<<<blk(NH), b256, 0, stream>>>(agg, NH);
            k_zero<<<blk(N), b256, 0, stream>>>(cnt, N);
            k_scatter_add<<<blk(E * H), b256, 0, stream>>>(ei[t], ei[t] + E, ib, agg, E, H);
            k_count<<<blk(E), b256, 0, stream>>>(ei[t] + E, cnt, E);
            k_divcnt<<<blk(NH), b256, 0, stream>>>(agg, cnt, N, H);
            gemm(ob, agg, Wn + (size_t)t * H * H, N, H, H, H, H, H, 1.0f, 0, 0);
        }
        if (relu_last) k_relu<<<blk(NH), b256, 0, stream>>>(ob, NH);
    };
    layer(hbuf, x, W1s, W1n, b1, 1);
    layer(zbuf, hbuf, W2s, W2n, b2, 0);

    // ---------------- fused distance + sigmoid -> sym ----------------
    k_rowq<<<blk(N), b256, 0, stream>>>(zbuf, lw, zwb, qv, N, H);
    gemm(symTm, zwb, zbuf, N, N, H, H, H, N, 0.0f, /*transB=*/1, 0);  // G = (z*w) @ z^T
    k_sym<<<blk(NN), b256, 0, stream>>>(symTm, qv, lb, N);

    // ---------------- Newton-Schulz matrix sqrt ----------------
    k_zero<<<1, 16, 0, stream>>>(sumsq, 16);
    k_sumsq<<<dim3(1024), b256, 0, stream>>>(symTm, sumsq, NN);
    k_initYZ<<<blk(NN), b256, 0, stream>>>(symTm, sumsq, Y0, Z0, N);

    float *Yc = Y0, *Yn = Y1, *Zc = Z0, *Zn = Z1;
    for (int it = 0; it < 6; ++it) {
        gemm(symTm, Zc, Yc, N, N, N, N, N, N, 0.0f, 0, 0);   // symTm = Z@Y
        k_tm<<<blk(NN), b256, 0, stream>>>(symTm, N);        // Tm = 0.5*(3I - Z@Y)
        gemm(Yn, Yc, symTm, N, N, N, N, N, N, 0.0f, 0, 0);   // Y' = Y@Tm
        gemm(Zn, symTm, Zc, N, N, N, N, N, N, 0.0f, 0, 0);   // Z' = Tm@Z
        float* t0 = Yc; Yc = Yn; Yn = t0;
        float* t1 = Zc; Zc = Zn; Zn = t1;
    }
    k_scaleS1<<<blk(NN), b256, 0, stream>>>(Yc, sumsq, o_S1, NN);

    // ---------------- outputs ----------------
    gemm(o_xnew, o_S1, x, N, H, N, N, H, H, 0.0f, 0, 0);     // x_new = S1 @ x

    float* adj = Z0;   // Z buffers free after NS
    float* tmp = Z1;
    for (int t = 0; t < 3; ++t) {
        k_zero<<<blk(NN), b256, 0, stream>>>(adj, NN);
        k_setedges<<<blk(E), b256, 0, stream>>>(ei[t], ei[t] + E, adj, N, E);
        gemm(tmp, o_S1, adj, N, N, N, N, N, N, 0.0f, 0, 0);  // S1 @ adj
        gemm(o_A[t], tmp, o_S1, N, N, N, N, N, N, 0.0f, 0, 0); // (S1@adj) @ S1
    }
}